// CRF_85100482003334
// MI455X (gfx1250) — compile-verified
//
#include <hip/hip_runtime.h>
#include <hip/hip_bf16.h>

// CRF NLL for B=512, S=1024, T=52 on gfx1250.
// Forward recurrence in the exp-domain: the 52x52 logsumexp contraction becomes
// an fp32 GEMM on V_WMMA_F32_16X16X4_F32 (16 batches x 52 tags, K=52 = 13x4).

typedef float v2f __attribute__((ext_vector_type(2)));
typedef float v8f __attribute__((ext_vector_type(8)));

#define CRF_B 512
#define CRF_S 1024
#define CRF_T 52                 // 50 targets + START + END
#define LDS_STRIDE 66            // row stride: j 0..63 storable; 8*66 % 64 = 16
                                 // -> half-wave store bank ranges disjoint
#define NEG_BIG (-1e30f)

// ---------------------------------------------------------------------------
// Gold path score + sequence lengths. Fully parallel over t (prev-tag is data,
// not a recurrence). One wave per batch. Runs FIRST so the forward kernel can
// use len[b] instead of per-step mask byte loads.
// ---------------------------------------------------------------------------
__global__ __launch_bounds__(256)
void crf_gold(const float* __restrict__ feats,
              const unsigned char* __restrict__ mask,
              const int* __restrict__ tags,
              const float* __restrict__ trans,
              float* __restrict__ gold_out,
              int* __restrict__ len_out)
{
    const int b    = blockIdx.x * (blockDim.x >> 5) + (threadIdx.x >> 5);
    const int lane = threadIdx.x & 31;
    if (b >= CRF_B) return;

    const long rowb = (long)b * CRF_S;
    float acc = 0.f;
    int   cnt = 0;
    for (int t = lane; t < CRF_S; t += 32) {
        if (mask[rowb + t]) {
            const int tag  = tags[rowb + t];
            const int prev = (t == 0) ? (CRF_T - 2) : tags[rowb + t - 1];
            acc += __builtin_nontemporal_load(&feats[(rowb + t) * CRF_T + tag])
                 + trans[prev * CRF_T + tag];
            ++cnt;
        }
    }
#pragma unroll
    for (int off = 16; off > 0; off >>= 1) {
        acc += __shfl_xor(acc, off, 32);
        cnt += __shfl_xor(cnt, off, 32);
    }
    if (lane == 0) {
        const int endtag = tags[rowb + (cnt - 1)];   // length >= 1 guaranteed
        gold_out[b] = acc + trans[endtag * CRF_T + (CRF_T - 1)];
        len_out[b]  = cnt;
    }
}

// ---------------------------------------------------------------------------
// Forward (log-partition) kernel: one wave32 per 16 batches, no barriers.
// part[] lives in registers in the WMMA C/D layout:
//   lane<16 -> batches r (0..7), lane>=16 -> batches r+8;  j = jt*16 + (lane&15)
// Per step: per-batch max (shfl butterflies) -> exp -> LDS transpose ->
// 4 chains x 13 fp32 WMMA -> log + emission + freeze-select.
// ---------------------------------------------------------------------------
__global__ __launch_bounds__(32)
void crf_forward_wmma(const float* __restrict__ feats,
                      const float* __restrict__ trans,
                      const int* __restrict__ len_in,
                      float* __restrict__ fwd_partial)
{
    __shared__ float aLDS[16 * LDS_STRIDE];

    const int lane   = threadIdx.x & 31;
    const int lanelo = lane & 15;
    const int half   = lane >> 4;      // 0: K in {4k,4k+1}, 1: K in {4k+2,4k+3}
    const int koff   = half * 2;
    const int b0     = blockIdx.x * 16;

    // ---- B operand: E = exp(transitions), resident in registers -----------
    // B-layout for 16x16x4 f32 WMMA: vgpr v, lanes0-15 -> (K=v+koff, N=lane)
    v2f Breg[4][13];
    float tEnd[4];                     // trans[j][END]
#pragma unroll
    for (int jt = 0; jt < 4; ++jt) {
        const int j      = jt * 16 + lanelo;
        const bool jv    = (j < CRF_T);
        const int jc     = jv ? j : (CRF_T - 1);
#pragma unroll
        for (int kk = 0; kk < 13; ++kk) {
            const int K0 = 4 * kk + koff;            // K0+1 <= 51 always
            const float e0 = __expf(trans[K0 * CRF_T + jc]);
            const float e1 = __expf(trans[(K0 + 1) * CRF_T + jc]);
            Breg[jt][kk].x = jv ? e0 : 0.f;
            Breg[jt][kk].y = jv ? e1 : 0.f;
        }
        tEnd[jt] = jv ? trans[jc * CRF_T + (CRF_T - 1)] : NEG_BIG;
    }

    // ---- per-batch lengths (prefix mask == t < len) ------------------------
    int lenr[8];
    int tmax = 1;
#pragma unroll
    for (int r = 0; r < 8; ++r) {
        lenr[r] = len_in[b0 + r + half * 8];
        tmax = max(tmax, lenr[r]);
    }
    tmax = max(tmax, __shfl_xor(tmax, 1, 32));
    tmax = max(tmax, __shfl_xor(tmax, 2, 32));
    tmax = max(tmax, __shfl_xor(tmax, 4, 32));
    tmax = max(tmax, __shfl_xor(tmax, 8, 32));
    tmax = max(tmax, __shfl_xor(tmax, 16, 32));      // wave-uniform loop bound

    // ---- init: part0[b][j] = feats[b,0,j] + trans[START][j] ---------------
    v8f part[4];
    const float* transStart = trans + (CRF_T - 2) * CRF_T;   // START row
#pragma unroll
    for (int jt = 0; jt < 4; ++jt) {
        const int j   = jt * 16 + lanelo;
        const bool jv = (j < CRF_T);
        const int jc  = jv ? j : (CRF_T - 1);
#pragma unroll
        for (int r = 0; r < 8; ++r) {
            const int m = r + half * 8;
            const float v =
                __builtin_nontemporal_load(
                    &feats[((long)(b0 + m) * CRF_S) * CRF_T + jc]) + transStart[jc];
            part[jt][r] = jv ? v : NEG_BIG;
        }
    }

    const v8f vzero = {0.f,0.f,0.f,0.f,0.f,0.f,0.f,0.f};

    for (int t = 1; t < tmax; ++t) {
        // ---- per-batch running max (16-lane half-wave butterflies) --------
        float mb[8];
#pragma unroll
        for (int r = 0; r < 8; ++r) {
            float mv = fmaxf(fmaxf(part[0][r], part[1][r]),
                             fmaxf(part[2][r], part[3][r]));
            mv = fmaxf(mv, __shfl_xor(mv, 1, 32));
            mv = fmaxf(mv, __shfl_xor(mv, 2, 32));
            mv = fmaxf(mv, __shfl_xor(mv, 4, 32));
            mv = fmaxf(mv, __shfl_xor(mv, 8, 32));
            mb[r] = mv;
        }

        // ---- a[b][j] = exp(part - max) -> LDS (unconditional; padded rows) -
#pragma unroll
        for (int jt = 0; jt < 4; ++jt) {
            const int j = jt * 16 + lanelo;
#pragma unroll
            for (int r = 0; r < 8; ++r) {
                const int m = r + half * 8;
                aLDS[m * LDS_STRIDE + j] = __expf(part[jt][r] - mb[r]);
            }
        }

        // ---- emission loads, branch-free (clamped col), issued early ------
        float fre[4][8];
#pragma unroll
        for (int r = 0; r < 8; ++r) {
            const long base = (long)(b0 + r + half * 8) * CRF_S + t;
#pragma unroll
            for (int jt = 0; jt < 4; ++jt) {
                const int j  = jt * 16 + lanelo;
                const int jc = (j < CRF_T) ? j : (CRF_T - 1);
                fre[jt][r] = __builtin_nontemporal_load(&feats[base * CRF_T + jc]);
            }
        }

        // ---- D = A x E : 4 independent chains of 13 fp32 WMMAs ------------
        // A-layout read: lane -> (M = lanelo batch row, K = 4kk + koff + {0,1})
        v8f acc[4];
#pragma unroll
        for (int jt = 0; jt < 4; ++jt) acc[jt] = vzero;

        const float* aRow = &aLDS[lanelo * LDS_STRIDE + koff];
#pragma unroll
        for (int kk = 0; kk < 13; ++kk) {
            const v2f a = *(const v2f*)(aRow + 4 * kk);   // ds_load_b64, 8B aligned
            acc[0] = __builtin_amdgcn_wmma_f32_16x16x4_f32(
                false, a, false, Breg[0][kk], (short)0, acc[0], false, false);
            acc[1] = __builtin_amdgcn_wmma_f32_16x16x4_f32(
                false, a, false, Breg[1][kk], (short)0, acc[1], false, false);
            acc[2] = __builtin_amdgcn_wmma_f32_16x16x4_f32(
                false, a, false, Breg[2][kk], (short)0, acc[2], false, false);
            acc[3] = __builtin_amdgcn_wmma_f32_16x16x4_f32(
                false, a, false, Breg[3][kk], (short)0, acc[3], false, false);
        }

        // ---- newpart = (t<len && j<T) ? log(D) + max + feats : oldpart ----
#pragma unroll
        for (int jt = 0; jt < 4; ++jt) {
            const int j   = jt * 16 + lanelo;
            const bool jv = (j < CRF_T);
#pragma unroll
            for (int r = 0; r < 8; ++r) {
                const float v    = acc[jt][r];
                const float cand =
                    (v > 0.f) ? (__logf(v) + mb[r] + fre[jt][r]) : NEG_BIG;
                const bool upd = jv && (t < lenr[r]);
                part[jt][r] = upd ? cand : part[jt][r];
            }
        }
    }

    // ---- final_b = logsumexp_i(part[b][i] + trans[i][END]); sum over 16 b -
    float bsum = 0.f;
#pragma unroll
    for (int r = 0; r < 8; ++r) {
        const float w0 = part[0][r] + tEnd[0];
        const float w1 = part[1][r] + tEnd[1];
        const float w2 = part[2][r] + tEnd[2];
        const float w3 = part[3][r] + tEnd[3];
        float mv = fmaxf(fmaxf(w0, w1), fmaxf(w2, w3));
        mv = fmaxf(mv, __shfl_xor(mv, 1, 32));
        mv = fmaxf(mv, __shfl_xor(mv, 2, 32));
        mv = fmaxf(mv, __shfl_xor(mv, 4, 32));
        mv = fmaxf(mv, __shfl_xor(mv, 8, 32));
        float s = __expf(w0 - mv) + __expf(w1 - mv) +
                  __expf(w2 - mv) + __expf(w3 - mv);
        s += __shfl_xor(s, 1, 32);
        s += __shfl_xor(s, 2, 32);
        s += __shfl_xor(s, 4, 32);
        s += __shfl_xor(s, 8, 32);
        bsum += mv + __logf(s);                    // final for batch r+half*8
    }
    bsum += __shfl_xor(bsum, 16, 32);              // combine the two halves
    if (lane == 0) fwd_partial[blockIdx.x] = bsum;
}

// ---------------------------------------------------------------------------
// Deterministic fixed-order reduction: out = sum(fwd) - sum(gold)
// ---------------------------------------------------------------------------
__global__ void crf_reduce(const float* __restrict__ fwd_partial,
                           const float* __restrict__ gold,
                           float* __restrict__ out)
{
    if (threadIdx.x == 0 && blockIdx.x == 0) {
        float f = 0.f;
        for (int i = 0; i < CRF_B / 16; ++i) f += fwd_partial[i];
        float g = 0.f;
        for (int i = 0; i < CRF_B; ++i) g += gold[i];
        out[0] = f - g;
    }
}

extern "C" void kernel_launch(void* const* d_in, const int* in_sizes, int n_in,
                              void* d_out, int out_size, void* d_ws, size_t ws_size,
                              hipStream_t stream) {
    (void)in_sizes; (void)n_in; (void)out_size; (void)ws_size;

    const float*         feats = (const float*)d_in[0];
    const unsigned char* mask  = (const unsigned char*)d_in[1];  // jax bool -> 1 byte
    const int*           tags  = (const int*)d_in[2];
    const float*         trans = (const float*)d_in[3];

    float* ws          = (float*)d_ws;
    float* fwd_partial = ws;                    // 32 floats
    float* gold        = ws + 32;               // 512 floats
    int*   lenArr      = (int*)(ws + 32 + CRF_B);  // 512 ints

    // 1) gold score + lengths (512 waves, 8 per block)
    crf_gold<<<CRF_B / 8, 256, 0, stream>>>(feats, mask, tags, trans, gold, lenArr);

    // 2) forward scan: 32 workgroups x 1 wave, each wave owns 16 batches
    crf_forward_wmma<<<CRF_B / 16, 32, 0, stream>>>(feats, trans, lenArr, fwd_partial);

    // 3) deterministic combine
    crf_reduce<<<1, 32, 0, stream>>>(fwd_partial, gold, (float*)d_out);
}